// SSM_54365696033203
// MI455X (gfx1250) — compile-verified
//
#include <hip/hip_runtime.h>

#define BATCH 64
#define LSEQ  8192
#define NDIM  64
#define TCH   128      // chunk length
#define NCH   64       // number of chunks = LSEQ / TCH

typedef __attribute__((ext_vector_type(2))) float v2f;
typedef __attribute__((ext_vector_type(4))) float v4f;
typedef __attribute__((ext_vector_type(8))) float v8f;
typedef __attribute__((ext_vector_type(4))) unsigned int u32x4;
typedef __attribute__((ext_vector_type(8))) unsigned int u32x8;

// ---------------- CDNA5 async global->LDS path (ASYNCcnt) ----------------
#if __has_builtin(__builtin_amdgcn_global_load_async_to_lds_b128)
#define HAVE_ASYNC 1
#else
#define HAVE_ASYNC 0
#endif

// builtin expects pointers to 16-byte int vectors (per hipcc diagnostic)
typedef int v4i_vs __attribute__((vector_size(4 * sizeof(int))));
typedef __attribute__((address_space(1))) v4i_vs gv4i;   // global
typedef __attribute__((address_space(3))) v4i_vs lv4i;   // LDS

__device__ __forceinline__ void copy16(void* lds_dst, const float* gsrc) {
#if HAVE_ASYNC
    __builtin_amdgcn_global_load_async_to_lds_b128((gv4i*)gsrc, (lv4i*)lds_dst, 0, 0);
#else
    *(v4f*)lds_dst = *(const v4f*)gsrc;
#endif
}

template <int N>
__device__ __forceinline__ void wait_async() {
#if HAVE_ASYNC
#if __has_builtin(__builtin_amdgcn_s_wait_asynccnt)
    __builtin_amdgcn_s_wait_asynccnt(N);
#else
    asm volatile("s_wait_asynccnt %0" ::"i"(N) : "memory");
#endif
#endif
}

// ---------------- CDNA5 Tensor Data Mover path (TENSORcnt) ----------------
#define USE_TDM 1

__device__ __forceinline__ unsigned lds_addr_of(const void* p) {
    return (unsigned)(unsigned long long)(__attribute__((address_space(3))) const void*)p;
}
__device__ __forceinline__ void wait_tensor0() {
#if __has_builtin(__builtin_amdgcn_s_wait_tensorcnt)
    __builtin_amdgcn_s_wait_tensorcnt(0);
#else
    asm volatile("s_wait_tensorcnt 0x0" ::: "memory");
#endif
}

// ---- WMMA f32 16x16x4 fragment helpers (wave32 layouts per CDNA5 ISA 7.12.2) ----
__device__ __forceinline__ v2f ldA(const float* p, int ld, int r0, int k0) {
    int l = threadIdx.x & 31, m = l & 15, h = l >> 4;
    const float* q = p + (r0 + m) * ld + k0 + 2 * h;
    v2f a; a.x = q[0]; a.y = q[1];
    return a;
}
__device__ __forceinline__ v2f ldB(const float* p, int ld, int k0, int c0) {
    int l = threadIdx.x & 31, n = l & 15, h = l >> 4;
    v2f b;
    b.x = p[(k0 + 2 * h + 0) * ld + c0 + n];
    b.y = p[(k0 + 2 * h + 1) * ld + c0 + n];
    return b;
}
__device__ __forceinline__ v8f ldC(const float* p, int ld, int r0, int c0) {
    int l = threadIdx.x & 31, n = l & 15, h = l >> 4;
    v8f c;
#pragma unroll
    for (int r = 0; r < 8; ++r) c[r] = p[(r0 + r + 8 * h) * ld + c0 + n];
    return c;
}
__device__ __forceinline__ void stD(float* p, int ld, int r0, int c0, v8f d) {
    int l = threadIdx.x & 31, n = l & 15, h = l >> 4;
#pragma unroll
    for (int r = 0; r < 8; ++r) p[(r0 + r + 8 * h) * ld + c0 + n] = d[r];
}

#define WMMA_F32(a, b, c) \
    __builtin_amdgcn_wmma_f32_16x16x4_f32(false, (a), false, (b), (short)0, (c), false, false)

// -------------------- Kernel 0: transpose u (BATCH x L) -> ut (L x BATCH) --------------------
__global__ __launch_bounds__(256) void k_transpose_u(const float* __restrict__ u,
                                                     float* __restrict__ ut) {
    __shared__ float tile[64][65];
    int t0 = blockIdx.x * 64;
    int tid = threadIdx.x;
    int g = tid >> 6;
    int c = tid & 63;
#pragma unroll
    for (int i = 0; i < 16; ++i) {
        int b = g * 16 + i;
        tile[b][c] = u[b * LSEQ + t0 + c];
    }
    __syncthreads();
#pragma unroll
    for (int i = 0; i < 16; ++i) {
        int tl = g * 16 + i;
        ut[(t0 + tl) * BATCH + c] = tile[c][tl];
    }
}

// -------------------- Kernel 1: setup (inverse, Abar, V', R, k, Ktri, P) --------------------
__global__ __launch_bounds__(256) void k_setup(const float* __restrict__ A,
                                               const float* __restrict__ Bv,
                                               const float* __restrict__ Cv,
                                               const float* __restrict__ stepp,
                                               float* __restrict__ Abar_g,
                                               float* __restrict__ P_g,
                                               float* __restrict__ Vp_g,
                                               float* __restrict__ R_g,
                                               float* __restrict__ Ktri_g) {
    __shared__ float aug[64][130];
    __shared__ float sN[64][65];
    __shared__ float prow[128];
    __shared__ float vv[64], vn[64], rr[64], rn[64], bs[64], cs[64];
    __shared__ float kk_s[TCH];

    int tid = threadIdx.x;
    float hs = stepp[0];
    float hh = 0.5f * hs;
    int row = tid & 63;
    int cg  = tid >> 6;

    for (int jj = 0; jj < 16; ++jj) {
        int j = cg * 16 + jj;
        float a   = A[row * 64 + j];
        float eye = (row == j) ? 1.0f : 0.0f;
        aug[row][j]      = eye - hh * a;
        aug[row][64 + j] = eye;
        sN[row][j]       = eye + hh * a;
    }
    __syncthreads();

    // Gauss-Jordan: matrix ~1.32*I - 0.005*U is strictly diagonally dominant -> no pivoting
    for (int c = 0; c < 64; ++c) {
        float pinv = 1.0f / aug[c][c];
        if (tid < 128) prow[tid] = aug[c][tid] * pinv;
        float f = aug[row][c];
        __syncthreads();
        if (row == c) {
            for (int jj = 0; jj < 32; ++jj) { int j = cg * 32 + jj; aug[c][j] = prow[j]; }
        } else {
            for (int jj = 0; jj < 32; ++jj) { int j = cg * 32 + jj; aug[row][j] -= f * prow[j]; }
        }
        __syncthreads();
    }
    // Abar = Minv @ (I + hh*A) -> left half
    for (int jj = 0; jj < 16; ++jj) {
        int j = cg * 16 + jj;
        float acc = 0.0f;
        for (int k = 0; k < 64; ++k) acc += aug[row][64 + k] * sN[k][j];
        aug[row][j] = acc;
        Abar_g[row * 64 + j] = acc;
    }
    if (tid < 64) {
        float acc = 0.0f;
        for (int k = 0; k < 64; ++k) acc += aug[tid][64 + k] * Bv[k];
        bs[tid] = hs * acc;
        cs[tid] = Cv[tid];
        vv[tid] = bs[tid];
        rr[tid] = cs[tid];
    }
    __syncthreads();

    for (int d = 0; d < TCH; ++d) {
        if (tid < 64) Vp_g[tid * TCH + (TCH - 1 - d)] = vv[tid];
        if (tid == 192) {
            float acc = 0.0f;
            for (int k = 0; k < 64; ++k) acc += cs[k] * vv[k];
            kk_s[d] = acc;
        }
        if (tid < 64) {
            float acc = 0.0f;
            for (int k = 0; k < 64; ++k) acc += aug[tid][k] * vv[k];
            vn[tid] = acc;
        } else if (tid < 128) {
            int j = tid - 64;
            float acc = 0.0f;
            for (int k = 0; k < 64; ++k) acc += rr[k] * aug[k][j];
            rn[j] = acc;
        }
        __syncthreads();
        if (tid < 64) {
            vv[tid] = vn[tid];
            rr[tid] = rn[tid];
            R_g[d * 64 + tid] = rn[tid];
        }
        __syncthreads();
    }

    for (int idx = tid; idx < TCH * TCH; idx += 256) {
        int i = idx >> 7, m = idx & 127;
        Ktri_g[idx] = (i >= m) ? kk_s[i - m] : 0.0f;
    }

    // P = Abar^TCH via 7 squarings
    for (int s = 0; s < 7; ++s) {
        __syncthreads();
        for (int jj = 0; jj < 16; ++jj) {
            int j = cg * 16 + jj;
            float acc = 0.0f;
            for (int k = 0; k < 64; ++k) acc += aug[row][k] * aug[k][j];
            aug[row][64 + j] = acc;
        }
        __syncthreads();
        for (int jj = 0; jj < 16; ++jj) {
            int j = cg * 16 + jj;
            aug[row][j] = aug[row][64 + j];
        }
    }
    __syncthreads();
    for (int jj = 0; jj < 16; ++jj) {
        int j = cg * 16 + jj;
        P_g[row * 64 + j] = aug[row][j];
    }
}

// ------------- Kernel 2: D_j = V'(64x128) @ U_j(128x64), async double-buffered -------------
__global__ __launch_bounds__(256) void k_chunk_gemm(const float* __restrict__ Vp,
                                                    const float* __restrict__ ut,
                                                    float* __restrict__ Dg) {
    __shared__ __align__(16) float As[2][64][20];   // 80B rows: 16B-aligned async targets
    __shared__ __align__(16) float Bs[2][16][64];
    int j = blockIdx.x;
    int tid = threadIdx.x;
    int w = tid >> 5;
    int t0 = 2 * w, t1 = 2 * w + 1;
    int ti0 = t0 >> 2, tj0 = t0 & 3, ti1 = t1 >> 2, tj1 = t1 & 3;
    v8f acc0 = {0.f,0.f,0.f,0.f,0.f,0.f,0.f,0.f};
    v8f acc1 = acc0;

    int ra = tid >> 2, qa = tid & 3;     // A slab: one b128 per thread
    int rb = tid >> 4, qb = tid & 15;    // B slab: one b128 per thread

    auto issue = [&](int kt, int buf) {
        int k0 = kt * 16;
        copy16(&As[buf][ra][qa * 4], Vp + ra * TCH + k0 + qa * 4);
        copy16(&Bs[buf][rb][qb * 4], ut + (j * TCH + k0 + rb) * BATCH + qb * 4);
    };

    const int NK = TCH / 16;             // 8 stages, 2 async instrs per wave per stage
    issue(0, 0);
    for (int kt = 0; kt < NK; ++kt) {
        int cur = kt & 1;
        if (kt + 1 < NK) {
            issue(kt + 1, cur ^ 1);
            wait_async<2>();             // oldest stage (kt) complete; kt+1 in flight
        } else {
            wait_async<0>();
        }
        __syncthreads();                 // all waves' stage-kt data visible
#pragma unroll
        for (int k4 = 0; k4 < 4; ++k4) {
            v2f a0 = ldA(&As[cur][0][0], 20, ti0 * 16, k4 * 4);
            v2f b0 = ldB(&Bs[cur][0][0], 64, k4 * 4, tj0 * 16);
            acc0 = WMMA_F32(a0, b0, acc0);
            v2f a1 = ldA(&As[cur][0][0], 20, ti1 * 16, k4 * 4);
            v2f b1 = ldB(&Bs[cur][0][0], 64, k4 * 4, tj1 * 16);
            acc1 = WMMA_F32(a1, b1, acc1);
        }
        __syncthreads();                 // done reading buf[cur]; may be overwritten next
    }
    float* D = Dg + j * (NDIM * BATCH);
    stD(D, BATCH, ti0 * 16, tj0 * 16, acc0);
    stD(D, BATCH, ti1 * 16, tj1 * 16, acc1);
}

// ---- Kernel 3: sequential combine s_{j+1} = P s_j + D_j ; P preloaded by the TDM ----
__global__ __launch_bounds__(256) void k_combine(const float* __restrict__ P_g,
                                                 const float* __restrict__ Dg,
                                                 float* __restrict__ Sg) {
    __shared__ __align__(16) float sP[64][65];
    __shared__ float sS[64][65];
    int tid = threadIdx.x;
    int w = tid >> 5;
    int t0 = 2 * w, t1 = t0 + 1;
    int ti0 = t0 >> 2, tj0 = t0 & 3, ti1 = t1 >> 2, tj1 = t1 & 3;

#if USE_TDM
    // TDM: 64x64 f32 tile, pad 1 dword every 64 dwords -> LDS leading dim 65 (bank-friendly)
    if (tid < 32) {   // one wave issues the descriptor (EXEC ignored by tensor ops)
        unsigned long long ga = (unsigned long long)P_g;
        u32x4 g0;
        g0.x = 1u;                                         // count = 1 valid descriptor
        g0.y = lds_addr_of(&sP[0][0]);                     // lds_addr
        g0.z = (unsigned)(ga & 0xffffffffu);               // global_addr[31:0]
        g0.w = (unsigned)((ga >> 32) & 0x1ffffffu) | (2u << 30);  // addr[56:32] | type=2
        u32x8 g1;
        g1.s0 = (2u << 16) | (1u << 20) | (5u << 22);      // 4B data, pad_en, 64dw interval, 1dw pad
        g1.s1 = (64u << 16);                               // tensor_dim0 = 64
        g1.s2 = (64u << 16);                               // tensor_dim1 = 64
        g1.s3 = (64u << 16);                               // tile_dim0   = 64
        g1.s4 = 64u;                                       // tile_dim1   = 64
        g1.s5 = 64u;                                       // tensor_dim0_stride = 64
        g1.s6 = 0u;
        g1.s7 = 0u;
        u32x4 gz = {0u, 0u, 0u, 0u};
        asm volatile("tensor_load_to_lds %0, %1, %2, %3"
                     :: "s"(g0), "s"(g1), "s"(gz), "s"(gz)
                     : "memory");
        wait_tensor0();
    }
#else
    for (int idx = tid; idx < 64 * 64; idx += 256) {
        sP[idx >> 6][idx & 63] = P_g[idx];
    }
#endif
    for (int idx = tid; idx < 64 * 64; idx += 256) {
        sS[idx >> 6][idx & 63] = 0.0f;
        Sg[idx] = 0.0f;                                    // s_0 = 0
    }
    __syncthreads();

    for (int j = 0; j < NCH; ++j) {
        const float* D = Dg + j * (NDIM * BATCH);
        v8f acc0 = ldC(D, BATCH, ti0 * 16, tj0 * 16);
        v8f acc1 = ldC(D, BATCH, ti1 * 16, tj1 * 16);
#pragma unroll
        for (int k4 = 0; k4 < 16; ++k4) {
            v2f a0 = ldA(&sP[0][0], 65, ti0 * 16, k4 * 4);
            v2f b0 = ldB(&sS[0][0], 65, k4 * 4, tj0 * 16);
            acc0 = WMMA_F32(a0, b0, acc0);
            v2f a1 = ldA(&sP[0][0], 65, ti1 * 16, k4 * 4);
            v2f b1 = ldB(&sS[0][0], 65, k4 * 4, tj1 * 16);
            acc1 = WMMA_F32(a1, b1, acc1);
        }
        __syncthreads();
        stD(&sS[0][0], 65, ti0 * 16, tj0 * 16, acc0);
        stD(&sS[0][0], 65, ti1 * 16, tj1 * 16, acc1);
        if (j + 1 < NCH) {
            float* Sn = Sg + (j + 1) * (NDIM * BATCH);
            stD(Sn, BATCH, ti0 * 16, tj0 * 16, acc0);
            stD(Sn, BATCH, ti1 * 16, tj1 * 16, acc1);
        }
        __syncthreads();
    }
}

// --- Kernel 4: Y_j = [R | Ktri] @ [s_j ; U_j] (128x192x64), async double-buffered ---
__global__ __launch_bounds__(256) void k_output(const float* __restrict__ R_g,
                                                const float* __restrict__ Ktri_g,
                                                const float* __restrict__ Sg,
                                                const float* __restrict__ ut,
                                                float* __restrict__ out) {
    __shared__ __align__(16) float As[2][128][20];
    __shared__ __align__(16) float Bs[2][16][64];
    int j = blockIdx.x;
    int tid = threadIdx.x;
    int w = tid >> 5;
    v8f zero8 = {0.f,0.f,0.f,0.f,0.f,0.f,0.f,0.f};
    v8f acc[4] = {zero8, zero8, zero8, zero8};
    const float* Sj = Sg + j * (NDIM * BATCH);

    int rb = tid >> 4, qb = tid & 15;

    auto issue = [&](int kt, int buf) {
        int k0 = kt * 16;                      // uniform; k0<64 boundary is slab-aligned
        const float* Abase; int Ald;
        const float* Bbase;
        if (k0 < 64) { Abase = R_g + k0;               Ald = 64;
                       Bbase = Sj + k0 * BATCH; }
        else         { Abase = Ktri_g + (k0 - 64);     Ald = TCH;
                       Bbase = ut + (j * TCH + (k0 - 64)) * BATCH; }
#pragma unroll
        for (int t = 0; t < 2; ++t) {          // A slab: 512 b128 over 256 threads
            int idx = tid + 256 * t;
            int r = idx >> 2, q = idx & 3;
            copy16(&As[buf][r][q * 4], Abase + r * Ald + q * 4);
        }
        copy16(&Bs[buf][rb][qb * 4], Bbase + rb * BATCH + qb * 4);
    };

    const int NK = 192 / 16;                   // 12 stages, 3 async instrs per wave per stage
    issue(0, 0);
    for (int kt = 0; kt < NK; ++kt) {
        int cur = kt & 1;
        if (kt + 1 < NK) {
            issue(kt + 1, cur ^ 1);
            wait_async<3>();
        } else {
            wait_async<0>();
        }
        __syncthreads();
#pragma unroll
        for (int k4 = 0; k4 < 4; ++k4) {
            v2f a = ldA(&As[cur][0][0], 20, w * 16, k4 * 4);
#pragma unroll
            for (int tj = 0; tj < 4; ++tj) {
                v2f b = ldB(&Bs[cur][0][0], 64, k4 * 4, tj * 16);
                acc[tj] = WMMA_F32(a, b, acc[tj]);
            }
        }
        __syncthreads();
    }
    // out[b][j*TCH + i] = Y[i][b]
    int l = tid & 31, n = l & 15, h = l >> 4;
#pragma unroll
    for (int tj = 0; tj < 4; ++tj) {
        int b = tj * 16 + n;
#pragma unroll
        for (int r = 0; r < 8; ++r) {
            int i = w * 16 + r + 8 * h;
            out[b * LSEQ + j * TCH + i] = acc[tj][r];
        }
    }
}

// ------------------------------------ launcher ------------------------------------
extern "C" void kernel_launch(void* const* d_in, const int* in_sizes, int n_in,
                              void* d_out, int out_size, void* d_ws, size_t ws_size,
                              hipStream_t stream) {
    (void)in_sizes; (void)n_in; (void)out_size; (void)ws_size;
    const float* u     = (const float*)d_in[0];   // (64, 8192)
    const float* A     = (const float*)d_in[1];   // (64, 64)
    const float* Bv    = (const float*)d_in[2];   // (64, 1)
    const float* Cv    = (const float*)d_in[3];   // (1, 64)
    const float* stepp = (const float*)d_in[4];   // scalar

    float* ws   = (float*)d_ws;
    float* Abar = ws;
    float* Pg   = Abar + 4096;
    float* Vp   = Pg   + 4096;
    float* Rg   = Vp   + 64 * TCH;
    float* Ktri = Rg   + TCH * 64;
    float* utb  = Ktri + TCH * TCH;
    float* Dg   = utb  + LSEQ * BATCH;
    float* Sg   = Dg   + NCH * NDIM * BATCH;

    k_transpose_u<<<LSEQ / 64, 256, 0, stream>>>(u, utb);
    k_setup<<<1, 256, 0, stream>>>(A, Bv, Cv, stepp, Abar, Pg, Vp, Rg, Ktri);
    k_chunk_gemm<<<NCH, 256, 0, stream>>>(Vp, utb, Dg);
    k_combine<<<1, 256, 0, stream>>>(Pg, Dg, Sg);
    k_output<<<NCH, 256, 0, stream>>>(Rg, Ktri, Sg, utb, (float*)d_out);
}